// DeepGCN_80401787781528
// MI455X (gfx1250) — compile-verified
//
#include <hip/hip_runtime.h>

#define NNODES 100000
#define NEDGES 1600000
#define NFEAT  128
#define NHID   32
#define NCLS   40

typedef float v2f __attribute__((ext_vector_type(2)));
typedef float v8f __attribute__((ext_vector_type(8)));

// ---------------------------------------------------------------------------
// Graph preprocessing (once per call): degree count, exclusive scan -> CSR
// rowptr by destination, fill (src, weight) pairs. deg[i] includes the self
// loop (+1); dinv = rsqrt(deg); edge weight = dinv[src]*dinv[dst].
// ---------------------------------------------------------------------------
__global__ void zero_u32_kernel(unsigned* __restrict__ p, int n) {
    int t = blockIdx.x * blockDim.x + threadIdx.x;
    if (t < n) p[t] = 0u;
}

__global__ void deg_count_kernel(const int* __restrict__ dst,
                                 unsigned* __restrict__ cnt, int E) {
    int e = blockIdx.x * blockDim.x + threadIdx.x;
    if (e < E) atomicAdd(&cnt[dst[e]], 1u);
}

__global__ void dinv_kernel(const unsigned* __restrict__ cnt,
                            float* __restrict__ dinv, int n) {
    int i = blockIdx.x * blockDim.x + threadIdx.x;
    if (i < n) dinv[i] = rsqrtf((float)cnt[i] + 1.0f);   // +1 = self loop
}

// Block-wise exclusive scan (256/block, Hillis-Steele in LDS) + block sums.
__global__ void scan_block_kernel(const unsigned* __restrict__ cnt,
                                  unsigned* __restrict__ excl,
                                  unsigned* __restrict__ bsum, int n) {
    __shared__ unsigned s[256];
    int t = blockIdx.x * 256 + threadIdx.x;
    unsigned v = (t < n) ? cnt[t] : 0u;
    s[threadIdx.x] = v;
    __syncthreads();
    for (int off = 1; off < 256; off <<= 1) {
        unsigned add = (threadIdx.x >= off) ? s[threadIdx.x - off] : 0u;
        __syncthreads();
        s[threadIdx.x] += add;
        __syncthreads();
    }
    if (t < n) excl[t] = s[threadIdx.x] - v;             // exclusive
    if (threadIdx.x == 255) bsum[blockIdx.x] = s[255];
}

// Serial exclusive scan of the (few hundred) block sums.
__global__ void scan_bsum_kernel(unsigned* __restrict__ bsum, int nb) {
    if (blockIdx.x == 0 && threadIdx.x == 0) {
        unsigned run = 0;
        for (int i = 0; i < nb; ++i) { unsigned v = bsum[i]; bsum[i] = run; run += v; }
    }
}

__global__ void scan_add_kernel(unsigned* __restrict__ excl,
                                const unsigned* __restrict__ bsum,
                                int n, unsigned total) {
    int t = blockIdx.x * blockDim.x + threadIdx.x;
    if (t < n) excl[t] += bsum[t >> 8];
    if (t == 0) excl[n] = total;                          // rowptr[N] = E
}

// Scatter edges into CSR buckets as packed (src, weight-bits) int2 pairs.
__global__ void csr_fill_kernel(const int* __restrict__ src, const int* __restrict__ dst,
                                const float* __restrict__ dinv,
                                const unsigned* __restrict__ rowptr,
                                unsigned* __restrict__ cursor,
                                int2* __restrict__ csr, int E) {
    int e = blockIdx.x * blockDim.x + threadIdx.x;
    if (e >= E) return;
    int s = src[e];
    int d = dst[e];
    unsigned pos = rowptr[d] + atomicAdd(&cursor[d], 1u);
    csr[pos] = make_int2(s, __float_as_int(dinv[s] * dinv[d]));
}

// ---------------------------------------------------------------------------
// Weight transpose + pad: Bt[n*K + k] = (n < Nout) ? W[k*Nout + n] : 0.
// Per-lane B fragments become contiguous b64 loads; padded columns produce
// garbage C columns that are simply never stored.
// ---------------------------------------------------------------------------
__global__ void transpose_pad_kernel(const float* __restrict__ W,
                                     float* __restrict__ Bt,
                                     int K, int Nout, int Npad) {
    int t = blockIdx.x * blockDim.x + threadIdx.x;
    if (t >= Npad * K) return;
    int n = t / K;
    int k = t - n * K;
    Bt[t] = (n < Nout) ? W[(size_t)k * Nout + n] : 0.0f;
}

// ---------------------------------------------------------------------------
// fp32 WMMA GEMM: C[M x Nout] = A[M x K] @ W[K x Nout], W pre-transposed as
// Bt[NT*16 x K]. One wave computes a 16 x (NT*16) strip; A fragment is loaded
// once per k-step and fed to NT independent V_WMMA_F32_16X16X4_F32.
// A frag (16x4 f32): lane row M = lane%16; reg r holds K = k0 + 2*(lane/16)+r.
// B frag (4x16 f32): lane col N = lane%16; same K indexing.
// C/D (16x16 f32): reg r -> row r + 8*(lane/16), col = lane%16.
// ---------------------------------------------------------------------------
template <int NT>
__global__ __launch_bounds__(256)
void gemm_wmma_kernel(const float* __restrict__ A, const float* __restrict__ Bt,
                      float* __restrict__ C, int M, int K, int Nout) {
    int wave = (int)((blockIdx.x * blockDim.x + threadIdx.x) >> 5);
    int lane = threadIdx.x & 31;
    if (wave >= (M >> 4)) return;              // wave-uniform: EXEC stays full

    int m0   = wave << 4;
    int half = lane >> 4;
    int l16  = lane & 15;

    const float* Arow = A + (size_t)(m0 + l16) * K;
    const float* Bcol[NT];
#pragma unroll
    for (int t = 0; t < NT; ++t)
        Bcol[t] = Bt + (size_t)(t * 16 + l16) * K;

    v8f c[NT] = {};

    for (int k0 = 0; k0 < K; k0 += 4) {
        int ka = k0 + (half << 1);
        v2f a = *(const v2f*)(Arow + ka);
#pragma unroll
        for (int t = 0; t < NT; ++t) {
            v2f b = *(const v2f*)(Bcol[t] + ka);
            c[t] = __builtin_amdgcn_wmma_f32_16x16x4_f32(
                       false, a, false, b, (short)0, c[t], false, false);
        }
    }

    int rbase = m0 + (half << 3);
#pragma unroll
    for (int t = 0; t < NT; ++t) {
        int ncol = t * 16 + l16;
        if (ncol < Nout) {
#pragma unroll
            for (int r = 0; r < 8; ++r)
                C[(size_t)(rbase + r) * Nout + ncol] = c[t][r];
        }
    }
}

// ---------------------------------------------------------------------------
// CSR gather-reduce fused with self-loop + bias, and (POST) with the layer
// epilogue  y = relu(agg) + x_old; x_old = y; x = y * (drop > 0.5 ? 2 : 0).
// One wave per node; lane = feature. Edge loop unrolled x2 so two independent
// 128B row gathers are in flight per wave. Zero atomics in the hot path.
// ---------------------------------------------------------------------------
template <int F, bool POST>
__global__ __launch_bounds__(256)
void gather_kernel(const float* __restrict__ h, const int2* __restrict__ csr,
                   const unsigned* __restrict__ rowptr,
                   const float* __restrict__ dinv, const float* __restrict__ bias,
                   const float* __restrict__ xold_in, const float* __restrict__ dropu,
                   float* __restrict__ xout, float* __restrict__ xold_out, int n) {
    int node = (int)((blockIdx.x * blockDim.x + threadIdx.x) >> 5);
    int lane = threadIdx.x & 31;
    if (node >= n) return;

    unsigned beg = rowptr[node];
    unsigned end = rowptr[node + 1];

    constexpr int C = (F + 31) / 32;            // 1 for F=32, 2 for F=40
    float acc[C];
    float di = dinv[node];
    float selfw = di * di;
    const float* hrow = h + (size_t)node * F;
#pragma unroll
    for (int c2 = 0; c2 < C; ++c2) {
        int f = lane + 32 * c2;
        acc[c2] = (f < F) ? hrow[f] * selfw + bias[f] : 0.0f;
    }

    unsigned j = beg;
    for (; j + 2 <= end; j += 2) {
        int2 p0 = csr[j];
        int2 p1 = csr[j + 1];
        const float* h0 = h + (size_t)p0.x * F;
        const float* h1 = h + (size_t)p1.x * F;
        float w0 = __int_as_float(p0.y);
        float w1 = __int_as_float(p1.y);
#pragma unroll
        for (int c2 = 0; c2 < C; ++c2) {
            int f = lane + 32 * c2;
            if (f < F) acc[c2] += h0[f] * w0 + h1[f] * w1;
        }
    }
    if (j < end) {
        int2 p0 = csr[j];
        const float* h0 = h + (size_t)p0.x * F;
        float w0 = __int_as_float(p0.y);
#pragma unroll
        for (int c2 = 0; c2 < C; ++c2) {
            int f = lane + 32 * c2;
            if (f < F) acc[c2] += h0[f] * w0;
        }
    }

    size_t base = (size_t)node * F;
#pragma unroll
    for (int c2 = 0; c2 < C; ++c2) {
        int f = lane + 32 * c2;
        if (f < F) {
            if (POST) {
                float y = fmaxf(acc[c2], 0.0f);
                if (xold_in) y += xold_in[base + f];
                xold_out[base + f] = y;
                xout[base + f] = y * (dropu[base + f] > 0.5f ? 2.0f : 0.0f);
            } else {
                xout[base + f] = acc[c2];
            }
        }
    }
}

// ---------------------------------------------------------------------------

extern "C" void kernel_launch(void* const* d_in, const int* in_sizes, int n_in,
                              void* d_out, int out_size, void* d_ws, size_t ws_size,
                              hipStream_t stream) {
    (void)in_sizes; (void)n_in; (void)out_size; (void)ws_size;

    const float* x0    = (const float*)d_in[0];   // [N, 128]
    const int*   ei    = (const int*)d_in[1];     // [2, E]
    const float* dropu = (const float*)d_in[2];   // [7, N, 32]
    const float* W0    = (const float*)d_in[3];   // [128, 32]
    const float* b0    = (const float*)d_in[4];   // [32]
    const float* Whid  = (const float*)d_in[5];   // [6, 32, 32]
    const float* bhid  = (const float*)d_in[6];   // [6, 32]
    const float* Wout  = (const float*)d_in[7];   // [32, 40]
    const float* bout  = (const float*)d_in[8];   // [40]
    float* out = (float*)d_out;                   // [N, 40]

    const int* src = ei;
    const int* dst = ei + NEDGES;

    // Workspace carve-out (256B aligned chunks)
    char* ws = (char*)d_ws;
    size_t off = 0;
    auto take = [&](size_t bytes) -> void* {
        void* p = (void*)(ws + off);
        off += (bytes + 255) & ~(size_t)255;
        return p;
    };
    const int NB = (NNODES + 255) / 256;                           // scan blocks
    unsigned* cnt    = (unsigned*)take((size_t)NNODES * 4);
    unsigned* rowptr = (unsigned*)take((size_t)(NNODES + 1) * 4);
    unsigned* cursor = (unsigned*)take((size_t)NNODES * 4);
    unsigned* bsum   = (unsigned*)take((size_t)NB * 4);
    float*    dinv   = (float*)   take((size_t)NNODES * 4);
    int2*     csr    = (int2*)    take((size_t)NEDGES * 8);        // (src, w)
    float*    h      = (float*)   take((size_t)NNODES * NCLS * 4); // max F = 40
    float*    xbuf   = (float*)   take((size_t)NNODES * NHID * 4);
    float*    xold   = (float*)   take((size_t)NNODES * NHID * 4);
    float*    Bt     = (float*)   take((size_t)48 * NFEAT * 4);    // padded W^T

    const int TB = 256;
    auto blocks = [](int n, int tb) { return (n + tb - 1) / tb; };

    // --- CSR build + normalization (once per call; capture-safe) ---
    zero_u32_kernel<<<blocks(NNODES, TB), TB, 0, stream>>>(cnt, NNODES);
    zero_u32_kernel<<<blocks(NNODES, TB), TB, 0, stream>>>(cursor, NNODES);
    deg_count_kernel<<<blocks(NEDGES, TB), TB, 0, stream>>>(dst, cnt, NEDGES);
    dinv_kernel<<<blocks(NNODES, TB), TB, 0, stream>>>(cnt, dinv, NNODES);
    scan_block_kernel<<<NB, 256, 0, stream>>>(cnt, rowptr, bsum, NNODES);
    scan_bsum_kernel<<<1, 32, 0, stream>>>(bsum, NB);
    scan_add_kernel<<<blocks(NNODES, TB), TB, 0, stream>>>(rowptr, bsum, NNODES, NEDGES);
    csr_fill_kernel<<<blocks(NEDGES, TB), TB, 0, stream>>>(
        src, dst, dinv, rowptr, cursor, csr, NEDGES);

    int gemm_blocks   = blocks(NNODES / 16, 8);    // 1 wave per 16-row strip
    int gather_blocks = blocks(NNODES, 8);         // 1 wave per node

    // --- 7 hidden GCNConv layers (GEMM -> fused gather+epilogue) ---
    const float* xold_in = nullptr;
    for (int layer = 0; layer < 7; ++layer) {
        const float* A = (layer == 0) ? x0 : xbuf;
        const float* W = (layer == 0) ? W0 : Whid + (size_t)(layer - 1) * NHID * NHID;
        const float* b = (layer == 0) ? b0 : bhid + (size_t)(layer - 1) * NHID;
        int K = (layer == 0) ? NFEAT : NHID;

        transpose_pad_kernel<<<blocks(NHID * K, TB), TB, 0, stream>>>(W, Bt, K, NHID, NHID);
        gemm_wmma_kernel<2><<<gemm_blocks, 256, 0, stream>>>(A, Bt, h, NNODES, K, NHID);

        gather_kernel<NHID, true><<<gather_blocks, 256, 0, stream>>>(
            h, csr, rowptr, dinv, b, xold_in,
            dropu + (size_t)layer * NNODES * NHID, xbuf, xold, NNODES);
        xold_in = xold;
    }

    // --- output GCNConv (gather directly into d_out, no epilogue) ---
    {
        transpose_pad_kernel<<<blocks(48 * NHID, TB), TB, 0, stream>>>(Wout, Bt, NHID, NCLS, 48);
        gemm_wmma_kernel<3><<<gemm_blocks, 256, 0, stream>>>(xbuf, Bt, h, NNODES, NHID, NCLS);

        gather_kernel<NCLS, false><<<gather_blocks, 256, 0, stream>>>(
            h, csr, rowptr, dinv, bout, nullptr, nullptr, out, nullptr, NNODES);
    }
}